// QuantumKernelMethod_65481071400434
// MI455X (gfx1250) — compile-verified
//
#include <hip/hip_runtime.h>

// Quantum kernel method, restructured:
//   k[i][j] = | <chi_j , psi_i> |^2
//   psi_i = 2 forward layers with angles x_i applied to |0>
//   chi_j = 2 adjoint layers (RZ(+y), reversed CNOT chain, RY(+y)) applied to |0>
// Phase 1: 128 waves evolve states fully in VGPRs (32 complex amps / lane).
// Phase 2: 16 waves compute the 64x64 complex Gram matrix with f32 WMMA.

#define NQ    10
#define DIM   1024         // 2^NQ
#define NX    64
#define NY    64
#define NREG  32           // DIM / 32 lanes
#define DEPTH 2

typedef __attribute__((ext_vector_type(2))) float v2f;
typedef __attribute__((ext_vector_type(8))) float v8f;

static __device__ __forceinline__ float lane_xor(float v, int mask) {
    return __shfl_xor(v, mask, 32);
}

// ---------------- gate primitives (all loops fully unrolled -> registers) ---

static __device__ __forceinline__ void ry_all(float (&sr)[NREG], float (&si)[NREG],
                                              const float (&c)[NQ], const float (&sn)[NQ],
                                              int lane) {
#pragma unroll
    for (int w = 0; w < NQ; ++w) {
        const int b = 9 - w;                   // wire w acts on bit (9-w)
        const float cc = c[w], ss = sn[w];
        if (b >= 5) {                          // register-bit butterfly
            const int m = 1 << (b - 5);
#pragma unroll
            for (int r = 0; r < NREG; ++r) {
                if (r & m) continue;
                const int r1 = r | m;
                const float ar = sr[r],  ai = si[r];
                const float br = sr[r1], bi = si[r1];
                sr[r]  = cc * ar - ss * br;  si[r]  = cc * ai - ss * bi;
                sr[r1] = ss * ar + cc * br;  si[r1] = ss * ai + cc * bi;
            }
        } else {                               // lane-bit butterfly via shuffle
            const int m = 1 << b;
            const float sg = (lane & m) ? ss : -ss;
#pragma unroll
            for (int r = 0; r < NREG; ++r) {
                const float orr = lane_xor(sr[r], m);
                const float oii = lane_xor(si[r], m);
                sr[r] = cc * sr[r] + sg * orr;
                si[r] = cc * si[r] + sg * oii;
            }
        }
    }
}

static __device__ __forceinline__ void rz_all(float (&sr)[NREG], float (&si)[NREG],
                                              const float (&c)[NQ], const float (&sn)[NQ],
                                              int lane) {
#pragma unroll
    for (int w = 0; w < NQ; ++w) {
        const int b = 9 - w;
        const float cc = c[w], ss = sn[w];
        if (b >= 5) {
            const int m = 1 << (b - 5);
#pragma unroll
            for (int r = 0; r < NREG; ++r) {
                const float sg = (r & m) ? ss : -ss;   // bit=1 -> e^{+i t/2}
                const float rr = sr[r], ii = si[r];
                sr[r] = cc * rr - sg * ii;
                si[r] = cc * ii + sg * rr;
            }
        } else {
            const int m = 1 << b;
            const float sg = (lane & m) ? ss : -ss;
#pragma unroll
            for (int r = 0; r < NREG; ++r) {
                const float rr = sr[r], ii = si[r];
                sr[r] = cc * rr - sg * ii;
                si[r] = cc * ii + sg * rr;
            }
        }
    }
}

// CNOT with control bit CB, target bit CB-1 (new[k] = old[k ^ tmask] iff ctrl set)
template <int CB>
static __device__ __forceinline__ void cnot_one(float (&sr)[NREG], float (&si)[NREG],
                                                int lane) {
    if constexpr (CB >= 6) {                   // both bits live in register index
        const int cm = 1 << (CB - 5);
        const int tm = 1 << (CB - 6);
#pragma unroll
        for (int r = 0; r < NREG; ++r) {
            if ((r & cm) && !(r & tm)) {
                const int r1 = r | tm;
                float t;
                t = sr[r]; sr[r] = sr[r1]; sr[r1] = t;
                t = si[r]; si[r] = si[r1]; si[r1] = t;
            }
        }
    } else if constexpr (CB == 5) {            // ctrl = reg bit0, target = lane bit4
#pragma unroll
        for (int r = 0; r < NREG; ++r) {
            if (r & 1) {
                sr[r] = lane_xor(sr[r], 16);
                si[r] = lane_xor(si[r], 16);
            }
        }
    } else {                                   // both bits live in lane index
        const int cm = 1 << CB;
        const int tm = 1 << (CB - 1);
        const bool ctl = (lane & cm) != 0;
#pragma unroll
        for (int r = 0; r < NREG; ++r) {
            const float orr = lane_xor(sr[r], tm);
            const float oii = lane_xor(si[r], tm);
            sr[r] = ctl ? orr : sr[r];
            si[r] = ctl ? oii : si[r];
        }
    }
}

static __device__ __forceinline__ void cnot_fwd(float (&sr)[NREG], float (&si)[NREG], int lane) {
    cnot_one<9>(sr, si, lane); cnot_one<8>(sr, si, lane); cnot_one<7>(sr, si, lane);
    cnot_one<6>(sr, si, lane); cnot_one<5>(sr, si, lane); cnot_one<4>(sr, si, lane);
    cnot_one<3>(sr, si, lane); cnot_one<2>(sr, si, lane); cnot_one<1>(sr, si, lane);
}
static __device__ __forceinline__ void cnot_rev(float (&sr)[NREG], float (&si)[NREG], int lane) {
    cnot_one<1>(sr, si, lane); cnot_one<2>(sr, si, lane); cnot_one<3>(sr, si, lane);
    cnot_one<4>(sr, si, lane); cnot_one<5>(sr, si, lane); cnot_one<6>(sr, si, lane);
    cnot_one<7>(sr, si, lane); cnot_one<8>(sr, si, lane); cnot_one<9>(sr, si, lane);
}

// ---------------- phase 1: state evolution, one wave per state ---------------
// ws layout (floats): psiR[64][1024] | psiI | chiR[64][1024] | chiI
#define PSI_R 0
#define PSI_I (64 * DIM)
#define CHI_R (2 * 64 * DIM)
#define CHI_I (3 * 64 * DIM)

__global__ __launch_bounds__(32) void qkm_evolve(const float* __restrict__ x,
                                                 const float* __restrict__ y,
                                                 float* __restrict__ ws) {
    const int s    = blockIdx.x;       // 0..63: psi(x_s)  64..127: chi(y_{s-64})
    const int lane = threadIdx.x;      // 0..31
    const bool adjoint = (s >= NX);
    const float* ang = adjoint ? (y + (size_t)(s - NX) * NQ) : (x + (size_t)s * NQ);

    float c[NQ], sn[NQ];
#pragma unroll
    for (int w = 0; w < NQ; ++w) {
        const float t = 0.5f * ang[w];
        c[w]  = cosf(t);
        sn[w] = sinf(t);
    }

    float sr[NREG], si[NREG];
#pragma unroll
    for (int r = 0; r < NREG; ++r) { sr[r] = 0.0f; si[r] = 0.0f; }
    if (lane == 0) sr[0] = 1.0f;       // |0...0>

    for (int d = 0; d < DEPTH; ++d) {
        if (!adjoint) {                // layer(a) = RZ * P * RY (RY acts first)
            ry_all(sr, si, c, sn, lane);
            cnot_fwd(sr, si, lane);
            rz_all(sr, si, c, sn, lane);
        } else {                       // layer(-y)^dagger = RY(y) * P_rev * RZ(y)
            rz_all(sr, si, c, sn, lane);
            cnot_rev(sr, si, lane);
            ry_all(sr, si, c, sn, lane);
        }
    }

    float* baseR;
    float* baseI;
    if (!adjoint) {
        baseR = ws + PSI_R + (size_t)s * DIM;
        baseI = ws + PSI_I + (size_t)s * DIM;
    } else {
        const int j = s - NX;
        baseR = ws + CHI_R + (size_t)j * DIM;
        baseI = ws + CHI_I + (size_t)j * DIM;
    }
#pragma unroll
    for (int r = 0; r < NREG; ++r) {   // k = r*32 + lane : fully coalesced
        baseR[r * 32 + lane] = sr[r];
        baseI[r * 32 + lane] = si[r];
    }
}

// ---------------- phase 2: complex Gram matrix via f32 WMMA ------------------
// out[i][j] = Re^2 + Im^2,  Re = sum psiR*chiR + psiI*chiI,
//                           Im = sum psiI*chiR - sum psiR*chiI
__global__ __launch_bounds__(32) void qkm_gram(const float* __restrict__ ws,
                                               float* __restrict__ out) {
    const int lane = threadIdx.x;
    const int ti   = blockIdx.x & 3;   // 4x4 tiles of 16x16
    const int tj   = blockIdx.x >> 2;
    const int i0   = ti * 16;
    const int j0   = tj * 16;

    const float* psiR = ws + PSI_R;
    const float* psiI = ws + PSI_I;
    const float* chiR = ws + CHI_R;
    const float* chiI = ws + CHI_I;

    // A 16x4 f32 fragment: lane = {m = lane&15, half = lane>>4}; VGPR v holds
    // K = 2*half + v. B 4x16 mirrors the split (lanes 0-15: K=0..1, 16-31: K=2..3).
    const int m    = lane & 15;
    const int half = lane >> 4;
    const int kk0  = half * 2;

    const float* aR = psiR + (size_t)(i0 + m) * DIM + kk0;
    const float* aI = psiI + (size_t)(i0 + m) * DIM + kk0;
    const float* bR = chiR + (size_t)(j0 + m) * DIM + kk0;
    const float* bI = chiI + (size_t)(j0 + m) * DIM + kk0;

    v8f acc_re = {0.f, 0.f, 0.f, 0.f, 0.f, 0.f, 0.f, 0.f};
    v8f acc_p  = {0.f, 0.f, 0.f, 0.f, 0.f, 0.f, 0.f, 0.f};
    v8f acc_m  = {0.f, 0.f, 0.f, 0.f, 0.f, 0.f, 0.f, 0.f};

#pragma unroll 4
    for (int k0 = 0; k0 < DIM; k0 += 4) {
        const v2f ar = *(const v2f*)(aR + k0);
        const v2f ai = *(const v2f*)(aI + k0);
        const v2f br = *(const v2f*)(bR + k0);
        const v2f bi = *(const v2f*)(bI + k0);
        // 8-arg form: (neg_a, A, neg_b, B, c_mod, C, reuse_a, reuse_b)
        acc_re = __builtin_amdgcn_wmma_f32_16x16x4_f32(false, ar, false, br,
                                                       (short)0, acc_re, false, false);
        acc_re = __builtin_amdgcn_wmma_f32_16x16x4_f32(false, ai, false, bi,
                                                       (short)0, acc_re, false, false);
        acc_p  = __builtin_amdgcn_wmma_f32_16x16x4_f32(false, ai, false, br,
                                                       (short)0, acc_p,  false, false);
        acc_m  = __builtin_amdgcn_wmma_f32_16x16x4_f32(false, ar, false, bi,
                                                       (short)0, acc_m,  false, false);
    }

    // C/D 16x16 f32 layout: VGPR v -> row (v + 8*half), col (lane & 15)
#pragma unroll
    for (int v = 0; v < 8; ++v) {
        const float re = acc_re[v];
        const float im = acc_p[v] - acc_m[v];
        const int   mi = v + 8 * half;
        const int   nj = lane & 15;
        out[(size_t)(i0 + mi) * NY + (j0 + nj)] = re * re + im * im;
    }
}

// ---------------- launch -----------------------------------------------------

extern "C" void kernel_launch(void* const* d_in, const int* in_sizes, int n_in,
                              void* d_out, int out_size, void* d_ws, size_t ws_size,
                              hipStream_t stream) {
    (void)in_sizes; (void)n_in; (void)out_size; (void)ws_size;
    const float* x = (const float*)d_in[0];   // (64, 10) f32
    const float* y = (const float*)d_in[1];   // (64, 10) f32
    float* out = (float*)d_out;               // (64, 64) f32
    float* ws  = (float*)d_ws;                // needs 4*64*1024 floats = 1 MiB

    qkm_evolve<<<dim3(2 * NX), dim3(32), 0, stream>>>(x, y, ws);
    qkm_gram  <<<dim3(16),     dim3(32), 0, stream>>>(ws, out);
}